// mLSTMCell_34522947125407
// MI455X (gfx1250) — compile-verified
//
#include <hip/hip_runtime.h>
#include <hip/hip_bf16.h>
#include <math.h>

#define BS     256
#define INPUT  1024
#define HIDDEN 2048
#define HEAD   128
#define EPS_LN 1e-5f
#define EPS_N  1e-6f

typedef __attribute__((ext_vector_type(2))) float        v2f;
typedef __attribute__((ext_vector_type(8))) float        v8f;
typedef __attribute__((ext_vector_type(4))) unsigned int v4u;
typedef __attribute__((ext_vector_type(4))) int          v4i;
typedef __attribute__((ext_vector_type(8))) int          v8i;

#if defined(__HIP_DEVICE_COMPILE__) && __has_builtin(__builtin_amdgcn_tensor_load_to_lds)
#define HAVE_TDM 1
#else
#define HAVE_TDM 0
#endif

__device__ __forceinline__ float sigmoidf_(float x) {
    return 1.0f / (1.0f + expf(-x));
}

// ---------------------------------------------------------------------------
// GEMM: Y[M=256, N] = X[256, K] @ W[N, K]^T + bias   (all f32, WMMA f32 16x16x4)
// Block = 256 threads = 8 waves. Block tile: 16(M) x 128(N). K-chunks of 64.
// X tile staged via vector loads; W tile (32 KB) staged via the Tensor Data
// Mover (TDM) with LDS padding (66-float row stride) done by the TDM pad unit.
// ---------------------------------------------------------------------------
#define TK 64
#define LDP (TK + 2)   // 66-float row stride (TDM: pad_interval=64 dw, pad_amount=2 dw)

__global__ __launch_bounds__(256) void gemm_xwT_kernel(
    const float* __restrict__ X, const float* __restrict__ W,
    const float* __restrict__ bias, float* __restrict__ Y,
    int N, int K)
{
    __shared__ float Xs[16][LDP];
    __shared__ float Ws[128][LDP];

    const int tid  = threadIdx.x;
    const int lane = tid & 31;
    const int wave = tid >> 5;           // 0..7
    const int m0   = blockIdx.y * 16;
    const int n0   = blockIdx.x * 128;

    v8f acc = {};

    const int am    = lane & 15;         // A-fragment M row
    const int khalf = (lane >> 4) * 2;   // K sub-offset per ISA 16x4 layout
    const int bn    = wave * 16 + (lane & 15);  // B-fragment N row (local)

    for (int k0 = 0; k0 < K; k0 += TK) {
        // ---- stage X tile: 16 x 64 floats, one float4 per thread ----
        {
            const int r  = tid >> 4;       // 0..15
            const int c4 = (tid & 15) * 4; // 0..60
            const float4 xv = *(const float4*)&X[(size_t)(m0 + r) * K + k0 + c4];
            Xs[r][c4 + 0] = xv.x; Xs[r][c4 + 1] = xv.y;
            Xs[r][c4 + 2] = xv.z; Xs[r][c4 + 3] = xv.w;
        }

#if HAVE_TDM
        // ---- stage W tile via TDM: 128 rows x 64 cols of f32 at (n0, k0) ----
        if (wave == 0) {
            const unsigned lds_base = (unsigned)(size_t)&Ws[0][0];
            const unsigned long long ga =
                (unsigned long long)(size_t)&W[(size_t)n0 * K + k0];
            v4u g0;
            g0.x = 1u;                                        // count=1 (valid user D#)
            g0.y = lds_base;                                  // lds_addr
            g0.z = (unsigned)(ga & 0xffffffffu);              // global_addr[31:0]
            g0.w = (unsigned)((ga >> 32) & 0x01ffffffu)       // global_addr[56:32]
                 | (2u << 30);                                // type=2 ("image")
            v8i g1;
            g1[0] = (int)((2u << 16)      // data_size=4B
                        | (1u << 20)      // pad_enable
                        | (5u << 22)      // pad_interval: 64 DWORDs (one 64-float row)
                        | (1u << 25));    // pad_amount: 2 DWORDs -> 66-float stride
            g1[1] = (int)(((unsigned)K & 0xffffu) << 16);     // tensor_dim0[15:0] @ bits63:48
            g1[2] = (int)(((unsigned)K >> 16)                 // tensor_dim0[31:16]
                        | (((unsigned)N & 0xffffu) << 16));   // tensor_dim1[15:0]
            g1[3] = (int)(((unsigned)N >> 16)                 // tensor_dim1[31:16]
                        | (64u << 16));                       // tile_dim0 = 64
            g1[4] = (int)128u;                                // tile_dim1 = 128, tile_dim2 = 0
            g1[5] = (int)K;                                   // tensor_dim0_stride[31:0]
            g1[6] = 0;                                        // stride hi | dim1_stride lo
            g1[7] = 0;
            const v4i gz = {0, 0, 0, 0};
#if defined(__clang_major__) && (__clang_major__ >= 23)
            const v8i gz8 = {0, 0, 0, 0, 0, 0, 0, 0};
            __builtin_amdgcn_tensor_load_to_lds(g0, g1, gz, gz, gz8, 0);
#else
            __builtin_amdgcn_tensor_load_to_lds(g0, g1, gz, gz, 0);
#endif
            __builtin_amdgcn_s_wait_tensorcnt(0);
        }
#else
        // ---- fallback: stage W tile with vector loads, 8 float4 per thread ----
        #pragma unroll
        for (int i = 0; i < 8; ++i) {
            const int idx = tid + i * 256;     // 0..2047
            const int r   = idx >> 4;          // 0..127
            const int c4  = (idx & 15) * 4;
            const float4 wv = *(const float4*)&W[(size_t)(n0 + r) * K + k0 + c4];
            Ws[r][c4 + 0] = wv.x; Ws[r][c4 + 1] = wv.y;
            Ws[r][c4 + 2] = wv.z; Ws[r][c4 + 3] = wv.w;
        }
#endif
        __syncthreads();

        // ---- 16 WMMA steps over this K-chunk ----
        #pragma unroll
        for (int kk = 0; kk < TK; kk += 4) {
            v2f a, b;
            a.x = Xs[am][kk + khalf];
            a.y = Xs[am][kk + khalf + 1];
            b.x = Ws[bn][kk + khalf];
            b.y = Ws[bn][kk + khalf + 1];
            acc = __builtin_amdgcn_wmma_f32_16x16x4_f32(
                false, a, false, b, (short)0, acc, false, false);
        }
        __syncthreads();
    }

    // ---- store: C/D layout -> VGPR r holds M = r + 8*(lane>>4), N = lane&15 ----
    const int col = n0 + wave * 16 + (lane & 15);
    const float bv = bias[col];
    #pragma unroll
    for (int r = 0; r < 8; ++r) {
        const int row = m0 + r + 8 * (lane >> 4);
        Y[(size_t)row * N + col] = acc[r] + bv;
    }
}

// ---------------------------------------------------------------------------
// k-layernorm + sum(k_ln^2): one 128-thread block per batch row
// ---------------------------------------------------------------------------
__global__ __launch_bounds__(128) void kln_kernel(
    const float* __restrict__ lin_k, const float* __restrict__ g,
    const float* __restrict__ be, float* __restrict__ k_ln,
    float* __restrict__ sumk2)
{
    __shared__ float red[128];
    const int b = blockIdx.x, t = threadIdx.x;
    const float x = lin_k[b * HEAD + t];

    red[t] = x; __syncthreads();
    for (int s = 64; s > 0; s >>= 1) { if (t < s) red[t] += red[t + s]; __syncthreads(); }
    const float mu = red[0] * (1.0f / HEAD);
    __syncthreads();

    const float d = x - mu;
    red[t] = d * d; __syncthreads();
    for (int s = 64; s > 0; s >>= 1) { if (t < s) red[t] += red[t + s]; __syncthreads(); }
    const float var = red[0] * (1.0f / HEAD);
    __syncthreads();

    const float kn = d * rsqrtf(var + EPS_LN) * g[t] + be[t];
    k_ln[b * HEAD + t] = kn;

    red[t] = kn * kn; __syncthreads();
    for (int s = 64; s > 0; s >>= 1) { if (t < s) red[t] += red[t + s]; __syncthreads(); }
    if (t == 0) sumk2[b] = red[0];
}

// ---------------------------------------------------------------------------
// v-layernorm row stats: one 256-thread block per batch row (over 2048)
// ---------------------------------------------------------------------------
__global__ __launch_bounds__(256) void vstats_kernel(
    const float* __restrict__ lin_v, float* __restrict__ mu_v,
    float* __restrict__ rinv_v)
{
    __shared__ float r1[256], r2[256];
    const int b = blockIdx.x, t = threadIdx.x;
    float s = 0.f, s2 = 0.f;
    for (int i = t; i < HIDDEN; i += 256) {
        const float x = lin_v[(size_t)b * HIDDEN + i];
        s += x; s2 += x * x;
    }
    r1[t] = s; r2[t] = s2; __syncthreads();
    for (int st = 128; st > 0; st >>= 1) {
        if (t < st) { r1[t] += r1[t + st]; r2[t] += r2[t + st]; }
        __syncthreads();
    }
    if (t == 0) {
        const float mu  = r1[0] * (1.0f / HIDDEN);
        const float var = r2[0] * (1.0f / HIDDEN) - mu * mu;
        mu_v[b]   = mu;
        rinv_v[b] = rsqrtf(var + EPS_LN);
    }
}

// ---------------------------------------------------------------------------
// Elementwise: gates, v-LN, n_new, h_new, and scale = i_exp * v_ln
// readout = scale * sum(k_ln^2)  (outer product collapsed analytically)
// ---------------------------------------------------------------------------
__global__ __launch_bounds__(256) void ew_kernel(
    const float* __restrict__ li, const float* __restrict__ lf,
    const float* __restrict__ lo, const float* __restrict__ lv,
    const float* __restrict__ mu_v, const float* __restrict__ rinv_v,
    const float* __restrict__ sumk2, const float* __restrict__ g,
    const float* __restrict__ be, float* __restrict__ out_h,
    float* __restrict__ out_n, float* __restrict__ scale)
{
    const int idx = blockIdx.x * 256 + threadIdx.x;   // [0, 256*2048)
    const int b = idx >> 11;
    const int h = idx & (HIDDEN - 1);

    const float ie = expf(sigmoidf_(li[idx]));
    const float fe = expf(sigmoidf_(lf[idx]));
    const float ov = sigmoidf_(lo[idx]);
    const float vl = (lv[idx] - mu_v[b]) * rinv_v[b] * g[h] + be[h];

    const float nn = fe + ie;
    const float sc = ie * vl;
    out_n[idx]  = nn;
    scale[idx]  = sc;
    out_h[idx]  = ov * (sc * sumk2[b]) / (nn + EPS_N);
}

// ---------------------------------------------------------------------------
// Stream C_new = scale[b,h] * k_ln[b, :]  -> 268 MB, float4-coalesced
// ---------------------------------------------------------------------------
__global__ __launch_bounds__(256) void cnew_kernel(
    const float* __restrict__ scale, const float* __restrict__ k_ln,
    float4* __restrict__ outC)
{
    const int i   = blockIdx.x * 256 + threadIdx.x;   // [0, 256*2048*32)
    const int row = i >> 5;          // b*2048 + h
    const int k4  = i & 31;          // which float4 of the 128-wide k row
    const float s = scale[row];
    const float4 kv = ((const float4*)k_ln)[(row >> 11) * 32 + k4];
    float4 o;
    o.x = s * kv.x; o.y = s * kv.y; o.z = s * kv.z; o.w = s * kv.w;
    outC[i] = o;
}

// ---------------------------------------------------------------------------
extern "C" void kernel_launch(void* const* d_in, const int* in_sizes, int n_in,
                              void* d_out, int out_size, void* d_ws, size_t ws_size,
                              hipStream_t stream) {
    (void)in_sizes; (void)n_in; (void)out_size; (void)ws_size;

    const float* x     = (const float*)d_in[0];
    const float* Wi_w  = (const float*)d_in[1];
    const float* Wi_b  = (const float*)d_in[2];
    const float* Wf_w  = (const float*)d_in[3];
    const float* Wf_b  = (const float*)d_in[4];
    const float* Wo_w  = (const float*)d_in[5];
    const float* Wo_b  = (const float*)d_in[6];
    // d_in[7], d_in[8]: Wq (dead code in reference output — skipped)
    const float* Wk_w  = (const float*)d_in[9];
    const float* Wk_b  = (const float*)d_in[10];
    const float* Wv_w  = (const float*)d_in[11];
    const float* Wv_b  = (const float*)d_in[12];
    const float* lnk_g = (const float*)d_in[15];
    const float* lnk_b = (const float*)d_in[16];
    const float* lnv_g = (const float*)d_in[17];
    const float* lnv_b = (const float*)d_in[18];

    // workspace layout (floats)
    float* ws     = (float*)d_ws;
    float* lin_i  = ws;                                   // 256*2048
    float* lin_f  = lin_i + (size_t)BS * HIDDEN;
    float* lin_o  = lin_f + (size_t)BS * HIDDEN;
    float* lin_v  = lin_o + (size_t)BS * HIDDEN;
    float* lin_k  = lin_v + (size_t)BS * HIDDEN;          // 256*128
    float* k_ln   = lin_k + (size_t)BS * HEAD;
    float* sumk2  = k_ln  + (size_t)BS * HEAD;            // 256
    float* mu_v   = sumk2 + BS;
    float* rinv_v = mu_v  + BS;
    float* scale  = rinv_v + BS;                          // 256*2048

    // output layout: h_new | C_new | n_new
    float* out_h = (float*)d_out;
    float* out_C = out_h + (size_t)BS * HIDDEN;
    float* out_n = out_C + (size_t)BS * HIDDEN * HEAD;

    const dim3 blk(256);
    const dim3 gH(HIDDEN / 128, BS / 16);   // (16,16)
    const dim3 gK(HEAD / 128, BS / 16);     // (1,16)

    gemm_xwT_kernel<<<gH, blk, 0, stream>>>(x, Wi_w, Wi_b, lin_i, HIDDEN, INPUT);
    gemm_xwT_kernel<<<gH, blk, 0, stream>>>(x, Wf_w, Wf_b, lin_f, HIDDEN, INPUT);
    gemm_xwT_kernel<<<gH, blk, 0, stream>>>(x, Wo_w, Wo_b, lin_o, HIDDEN, INPUT);
    gemm_xwT_kernel<<<gH, blk, 0, stream>>>(x, Wv_w, Wv_b, lin_v, HIDDEN, INPUT);
    gemm_xwT_kernel<<<gK, blk, 0, stream>>>(x, Wk_w, Wk_b, lin_k, HEAD,   INPUT);

    kln_kernel<<<BS, 128, 0, stream>>>(lin_k, lnk_g, lnk_b, k_ln, sumk2);
    vstats_kernel<<<BS, 256, 0, stream>>>(lin_v, mu_v, rinv_v);

    ew_kernel<<<(BS * HIDDEN) / 256, blk, 0, stream>>>(
        lin_i, lin_f, lin_o, lin_v, mu_v, rinv_v, sumk2, lnv_g, lnv_b,
        out_h, out_n, scale);

    cnew_kernel<<<(BS * HIDDEN * HEAD / 4) / 256, blk, 0, stream>>>(
        scale, k_ln, (float4*)out_C);
}